// AQLMQuantizedLinear_5781025980673
// MI455X (gfx1250) — compile-verified
//
#include <hip/hip_runtime.h>
#include <stdint.h>

// AQLM 1x16 dequant-fused GEMM for MI455X (gfx1250, wave32, WMMA f32 16x16x4).
// out[t,o] = scale[o] * sum_ig sum_j x[t, ig*8+j] * cb[code[o,ig]][j] + bias[o]

typedef __attribute__((ext_vector_type(2))) float v2f;
typedef __attribute__((ext_vector_type(4))) float v4f;
typedef __attribute__((ext_vector_type(8))) float v8f;

#define IN_F       4096
#define OUT_F      11008
#define TOKENS     64
#define N_IG       512      // IN_F / 8
#define K_CHUNK    256      // K elements staged in LDS per iteration
#define IG_CHUNK   32       // K_CHUNK / 8 AQLM groups per chunk
#define LDS_STRIDE 260      // padded row stride (floats) -> conflict-free ds_load_b64

__global__ __launch_bounds__(256) void aqlm_wmma_f32_kernel(
    const float* __restrict__ x,          // [64, 4096]
    const int*   __restrict__ codes,      // [11008, 512]
    const float* __restrict__ codebooks,  // [65536, 8]
    const float* __restrict__ scales,     // [11008]
    const float* __restrict__ bias,       // [11008]
    float*       __restrict__ out)        // [64, 11008]
{
    __shared__ float xs[TOKENS * LDS_STRIDE];

    const int tid   = threadIdx.x;
    const int lane  = tid & 31;
    const int wave  = tid >> 5;
    const int n0    = blockIdx.x * 128 + wave * 16;   // 16 output features per wave

    const int nlane = lane & 15;        // N index within tile (and token-in-tile for A)
    const int hi    = lane >> 4;        // lane half selects K sub-pair {0,1} vs {2,3}
    const int off2  = hi * 2;

    const int      nfeat    = n0 + nlane;
    // 32-bit byte offsets so loads select the SADDR (uniform base + u32 voffset) form.
    const uint32_t codeBase = (uint32_t)nfeat * (uint32_t)N_IG;   // element offset into codes
    const uint32_t laneSub  = (uint32_t)off2 * 4u;                // byte sub-offset in cb row
    const char*    cbBase   = (const char*)codebooks;

    v8f acc[4];
    #pragma unroll
    for (int m = 0; m < 4; ++m) acc[m] = (v8f){};

    for (int kc = 0; kc < IN_F / K_CHUNK; ++kc) {
        __syncthreads();
        // Cooperative coalesced stage of x[0..63][kc*256 .. kc*256+255] into LDS.
        {
            const float* xsrc = x + (size_t)kc * K_CHUNK;
            #pragma unroll
            for (int i = 0; i < 16; ++i) {
                int idx = tid + i * 256;      // float4 slot: 4096 slots total
                int t   = idx >> 6;           // token row (64 float4 per row)
                int c4  = idx & 63;
                v4f v = *(const v4f*)(xsrc + (size_t)t * IN_F + c4 * 4);
                *(v4f*)(&xs[t * LDS_STRIDE + c4 * 4]) = v;
            }
        }
        __syncthreads();

        const int igbase = kc * IG_CHUNK;
        // Prefetch next chunk of this feature's codes row (global_prefetch_b8).
        if (kc + 1 < IN_F / K_CHUNK)
            __builtin_prefetch(codes + codeBase + (uint32_t)(igbase + IG_CHUNK), 0, 1);

        #pragma unroll 4
        for (int igl = 0; igl < IG_CHUNK; ++igl) {
            // --- B operand: gather dequant weights for 16 output features ---
            // lanes L and L+16 load the same code (same cache line, broadcast).
            uint32_t code = (uint32_t)codes[codeBase + (uint32_t)(igbase + igl)];
            uint32_t boff = (code << 5) + laneSub;              // one v_lshl_add_u32
            v2f b1 = *(const v2f*)(cbBase + boff);              // K = off2, off2+1
            v2f b2 = *(const v2f*)(cbBase + boff + 16);         // K = off2+4, off2+5

            const int kk = igl * 8 + off2;
            // --- 4 M-tiles of 16 tokens, 2 WMMA K-steps each ---
            #pragma unroll
            for (int m = 0; m < 4; ++m) {
                const float* ax = &xs[(size_t)(m * 16 + nlane) * LDS_STRIDE + kk];
                v2f a1 = *(const v2f*)(ax);
                v2f a2 = *(const v2f*)(ax + 4);
                acc[m] = __builtin_amdgcn_wmma_f32_16x16x4_f32(
                    false, a1, false, b1, (short)0, acc[m], false, false);
                acc[m] = __builtin_amdgcn_wmma_f32_16x16x4_f32(
                    false, a2, false, b2, (short)0, acc[m], false, false);
            }
        }
    }

    // Epilogue: fold per-feature scale, add bias.
    const float s  = scales[nfeat];
    const float bv = bias[nfeat];
    #pragma unroll
    for (int m = 0; m < 4; ++m) {
        #pragma unroll
        for (int v = 0; v < 8; ++v) {
            int token = m * 16 + hi * 8 + v;   // C/D layout: VGPR v -> M=v (+8 hi half)
            out[(size_t)token * OUT_F + nfeat] = acc[m][v] * s + bv;
        }
    }
}

extern "C" void kernel_launch(void* const* d_in, const int* in_sizes, int n_in,
                              void* d_out, int out_size, void* d_ws, size_t ws_size,
                              hipStream_t stream) {
    const float* x         = (const float*)d_in[0];
    const int*   codes     = (const int*)d_in[1];
    const float* codebooks = (const float*)d_in[2];
    const float* scales    = (const float*)d_in[3];
    const float* bias      = (const float*)d_in[4];
    float*       out       = (float*)d_out;

    dim3 grid(OUT_F / 128);   // 86 blocks, 8 waves each, 16 features per wave
    dim3 block(256);
    aqlm_wmma_f32_kernel<<<grid, block, 0, stream>>>(x, codes, codebooks, scales, bias, out);
}